// GIN_12816182411598
// MI455X (gfx1250) — compile-verified
//
#include <hip/hip_runtime.h>
#include <hip/hip_bf16.h>

typedef float v2f __attribute__((ext_vector_type(2)));
typedef float v8f __attribute__((ext_vector_type(8)));

#define DIM 64
#define EDIM 16
#define NG 2048
#define BN_EPS 1e-5f
#define ASTRIDE 260

__device__ __forceinline__ float lrelu_f(float v) { return v >= 0.0f ? v : 0.01f * v; }

// D = A(16x4 f32) * B(4x16 f32) + C(16x16 f32), wave32 WMMA
__device__ __forceinline__ v8f wmma_f32(v2f a, v2f b, v8f c) {
    return __builtin_amdgcn_wmma_f32_16x16x4_f32(false, a, false, b, (short)0, c, false, false);
}

// ---------------------------------------------------------------------------
// Edge phase: e = edge_attr @ We^T + be ; m = relu(x[src] + e) ; agg[dst] += m
// One wave handles a tile of 16 edges; 4 N-tiles x 4 K-steps = 16 WMMAs/tile.
// 32-bit element offsets (saddr-form globals); gathers batched before atomics.
// ---------------------------------------------------------------------------
__global__ __launch_bounds__(256) void gine_edge_kernel(
    const float* __restrict__ x, const int* __restrict__ src, const int* __restrict__ dst,
    const float* __restrict__ eattr, const float* __restrict__ We, const float* __restrict__ be,
    float* __restrict__ agg, int nEdges)
{
    const int lane = threadIdx.x & 31;
    const int wave = threadIdx.x >> 5;
    const int half = lane >> 4;
    const int row  = lane & 15;
    const int tile = blockIdx.x * 8 + wave;
    const int base = tile * 16;
    if (base >= nEdges) return;

    if (base + 16 <= nEdges) {
        // ---- full-tile fast path (always taken for E % 16 == 0) ----
        const unsigned e0 = (unsigned)base + 8u * half;   // 32B-aligned
        const int4 s0 = *(const int4*)(src + e0);
        const int4 s1 = *(const int4*)(src + e0 + 4);
        const int4 d0 = *(const int4*)(dst + e0);
        const int4 d1 = *(const int4*)(dst + e0 + 4);
        const int sn[8] = {s0.x, s0.y, s0.z, s0.w, s1.x, s1.y, s1.z, s1.w};
        const int dn[8] = {d0.x, d0.y, d0.z, d0.w, d1.x, d1.y, d1.z, d1.w};
        unsigned xi[8], ai[8];
#pragma unroll
        for (int v = 0; v < 8; ++v) {
            xi[v] = (unsigned)sn[v] * DIM;
            ai[v] = (unsigned)dn[v] * DIM;
        }

        // A fragments: edge_attr tile [16 edges x 16 attrs], 4 K-steps of 4
        v2f a[4];
        {
            const float* ar = eattr + (unsigned)(base + row) * EDIM;
#pragma unroll
            for (int ks = 0; ks < 4; ++ks)
                a[ks] = *(const v2f*)(ar + ks * 4 + half * 2);
        }

#pragma unroll
        for (int nt = 0; nt < 4; ++nt) {
            const unsigned n = nt * 16 + row;             // output dim owned by this lane
            float bias = be[n];
            v8f c;
#pragma unroll
            for (int i = 0; i < 8; ++i) c[i] = bias;
#pragma unroll
            for (int ks = 0; ks < 4; ++ks) {
                v2f bf = *(const v2f*)(We + n * EDIM + ks * 4 + half * 2); // B[k][n]=We[n][k]
                c = wmma_f32(a[ks], bf, c);
            }
            // c[v] = element (edge row v+8*half, dim n)
            // (a) issue all 8 gathers so they overlap ...
            float xv[8];
#pragma unroll
            for (int v = 0; v < 8; ++v) xv[v] = x[xi[v] + n];
            // (b) ... then relu + scatter-add
#pragma unroll
            for (int v = 0; v < 8; ++v) {
                float m = fmaxf(c[v] + xv[v], 0.0f);
                atomicAdd(&agg[ai[v] + n], m);
            }
        }
    } else {
        // ---- scalar tail (never taken for the harness sizes) ----
        for (int e = base; e < nEdges; ++e) {
            const int s = src[e], d = dst[e];
            for (int dd = lane; dd < DIM; dd += 32) {
                float acc = be[dd];
                for (int k = 0; k < EDIM; ++k)
                    acc += eattr[(unsigned)e * EDIM + k] * We[dd * EDIM + k];
                float m = acc + x[(unsigned)s * DIM + dd];
                m = fmaxf(m, 0.0f);
                atomicAdd(&agg[(unsigned)d * DIM + dd], m);
            }
        }
    }
}

// ---------------------------------------------------------------------------
// Node phase: z = (x + agg) @ W^T + b ; BatchNorm(eval) ; LeakyReLU -> h
// One wave per 16-node tile; 4 N-tiles x 16 K-steps = 64 WMMAs/tile.
// ---------------------------------------------------------------------------
__global__ __launch_bounds__(256) void gine_node_kernel(
    const float* __restrict__ x, const float* __restrict__ agg,
    const float* __restrict__ W, const float* __restrict__ b,
    const float* __restrict__ g, const float* __restrict__ bt,
    const float* __restrict__ mean, const float* __restrict__ var,
    float* __restrict__ h, int nNodes)
{
    const int lane = threadIdx.x & 31;
    const int wave = threadIdx.x >> 5;
    const int half = lane >> 4;
    const int row  = lane & 15;
    const int base = (blockIdx.x * 8 + wave) * 16;
    if (base >= nNodes) return;

    if (base + 16 <= nNodes) {
        v2f a[16];
        {
            const unsigned ro = (unsigned)(base + row) * DIM;
#pragma unroll
            for (int ks = 0; ks < 16; ++ks) {
                unsigned k = ro + ks * 4 + half * 2;
                v2f xv = *(const v2f*)(x + k);
                v2f gv = *(const v2f*)(agg + k);
                a[ks].x = xv.x + gv.x;
                a[ks].y = xv.y + gv.y;
            }
        }

#pragma unroll
        for (int nt = 0; nt < 4; ++nt) {
            const unsigned n = nt * 16 + row;
            float bias = b[n];
            v8f c;
#pragma unroll
            for (int i = 0; i < 8; ++i) c[i] = bias;
#pragma unroll
            for (int ks = 0; ks < 16; ++ks) {
                v2f bf = *(const v2f*)(W + n * DIM + ks * 4 + half * 2); // B[k][n]=W[n][k]
                c = wmma_f32(a[ks], bf, c);
            }
            float scale = g[n] * rsqrtf(var[n] + BN_EPS);
            float shift = bt[n] - mean[n] * scale;
#pragma unroll
            for (int v = 0; v < 8; ++v) {
                unsigned node = (unsigned)base + v + 8 * half;
                float z = c[v] * scale + shift;
                h[node * DIM + n] = lrelu_f(z);
            }
        }
    } else {
        for (int node = base; node < nNodes; ++node) {
            for (int dd = lane; dd < DIM; dd += 32) {
                float acc = b[dd];
                for (int k = 0; k < DIM; ++k)
                    acc += (x[(unsigned)node * DIM + k] + agg[(unsigned)node * DIM + k]) * W[dd * DIM + k];
                float scale = g[dd] * rsqrtf(var[dd] + BN_EPS);
                float z = (acc - mean[dd]) * scale + bt[dd];
                h[(unsigned)node * DIM + dd] = lrelu_f(z);
            }
        }
    }
}

__global__ void zero_kernel(float* __restrict__ p, long n) {
    long i = (long)blockIdx.x * blockDim.x + threadIdx.x;
    if (i < n) p[i] = 0.0f;
}

__global__ void pool_kernel(const float* __restrict__ h3, const int* __restrict__ batch,
                            float* __restrict__ hpool, int nNodes) {
    unsigned i = blockIdx.x * blockDim.x + threadIdx.x;
    unsigned total = (unsigned)nNodes * DIM;
    if (i < total) {
        unsigned node = i >> 6;
        unsigned dim  = i & 63;
        atomicAdd(&hpool[(unsigned)batch[node] * DIM + dim], h3[i]);
    }
}

// ---------------------------------------------------------------------------
// Head: h = concat(h1,h2,h3,hpool[batch]) [N,256]; o1 = lrelu(h@Wl1^T + bl1);
// out = o1 @ Wl2^T + bl2. The 256->1 dot is fused: o1 never hits HBM.
// One wave per 16-node tile; A row staged in padded LDS; 16x64 WMMAs/tile.
// ---------------------------------------------------------------------------
__global__ __launch_bounds__(64) void head_kernel(
    const float* __restrict__ h1, const float* __restrict__ h2, const float* __restrict__ h3,
    const float* __restrict__ hpool, const int* __restrict__ batch,
    const float* __restrict__ Wl1, const float* __restrict__ bl1,
    const float* __restrict__ Wl2, const float* __restrict__ bl2,
    float* __restrict__ out, int nNodes)
{
    __shared__ float At[2][16][ASTRIDE];
    const int lane = threadIdx.x & 31;
    const int wave = threadIdx.x >> 5;
    const int half = lane >> 4;
    const int row  = lane & 15;
    const int base = (blockIdx.x * 2 + wave) * 16;
    if (base >= nNodes) return;

    float (*A)[ASTRIDE] = At[wave];
    for (int idx = lane; idx < 16 * 256; idx += 32) {
        int r = idx >> 8;
        int k = idx & 255;
        int node = base + r;
        if (node >= nNodes) node = nNodes - 1;
        unsigned no = (unsigned)node * DIM;
        float v;
        if (k < 64)       v = h1[no + k];
        else if (k < 128) v = h2[no + (k - 64)];
        else if (k < 192) v = h3[no + (k - 128)];
        else              v = hpool[(unsigned)batch[node] * DIM + (k - 192)];
        A[r][k] = v;
    }
    // wave-local LDS RAW fence (CDNA5 split counters)
    asm volatile("s_wait_dscnt 0x0" ::: "memory");

    float racc[8];
#pragma unroll
    for (int v = 0; v < 8; ++v) racc[v] = 0.0f;

    for (int nt = 0; nt < 16; ++nt) {
        const unsigned n = nt * 16 + row;
        // prefetch next N-tile's B block while this one computes
        if (nt < 15) __builtin_prefetch(Wl1 + (n + 16) * 256 + half * 128, 0, 0);
        float bias = bl1[n];
        v8f c;
#pragma unroll
        for (int i = 0; i < 8; ++i) c[i] = bias;
#pragma unroll
        for (int ks = 0; ks < 64; ++ks) {
            int k = ks * 4 + half * 2;
            v2f af, bf;
            af.x = A[row][k];
            af.y = A[row][k + 1];
            bf = *(const v2f*)(Wl1 + n * 256 + k);        // B[k][n] = Wl1[n][k]
            c = wmma_f32(af, bf, c);
        }
        float w2 = Wl2[n];
#pragma unroll
        for (int v = 0; v < 8; ++v) racc[v] += lrelu_f(c[v]) * w2;
    }

    // reduce the 16 lanes inside each half-wave (xor masks stay within halves)
#pragma unroll
    for (int m = 1; m <= 8; m <<= 1) {
#pragma unroll
        for (int v = 0; v < 8; ++v) racc[v] += __shfl_xor(racc[v], m, 32);
    }
    if (row == 0) {
        float bias2 = bl2[0];
#pragma unroll
        for (int v = 0; v < 8; ++v) {
            int node = base + v + 8 * half;
            if (node < nNodes) out[node] = racc[v] + bias2;
        }
    }
}

extern "C" void kernel_launch(void* const* d_in, const int* in_sizes, int n_in,
                              void* d_out, int out_size, void* d_ws, size_t ws_size,
                              hipStream_t stream)
{
    (void)n_in; (void)out_size; (void)ws_size;

    const float* x     = (const float*)d_in[0];
    const int*   ei    = (const int*)d_in[1];
    const float* eattr = (const float*)d_in[2];
    const int*   batch = (const int*)d_in[3];

    const float* We[3] = {(const float*)d_in[4],  (const float*)d_in[12], (const float*)d_in[20]};
    const float* be[3] = {(const float*)d_in[5],  (const float*)d_in[13], (const float*)d_in[21]};
    const float* W[3]  = {(const float*)d_in[6],  (const float*)d_in[14], (const float*)d_in[22]};
    const float* b[3]  = {(const float*)d_in[7],  (const float*)d_in[15], (const float*)d_in[23]};
    const float* g[3]  = {(const float*)d_in[8],  (const float*)d_in[16], (const float*)d_in[24]};
    const float* bt[3] = {(const float*)d_in[9],  (const float*)d_in[17], (const float*)d_in[25]};
    const float* mn[3] = {(const float*)d_in[10], (const float*)d_in[18], (const float*)d_in[26]};
    const float* vr[3] = {(const float*)d_in[11], (const float*)d_in[19], (const float*)d_in[27]};
    const float* Wl1 = (const float*)d_in[28];
    const float* bl1 = (const float*)d_in[29];
    const float* Wl2 = (const float*)d_in[30];
    const float* bl2 = (const float*)d_in[31];

    const int N = in_sizes[0] / DIM;
    const int E = in_sizes[2] / EDIM;
    const int* src = ei;
    const int* dst = ei + E;

    float* agg   = (float*)d_ws;
    float* h1    = agg + (size_t)N * DIM;
    float* h2    = h1  + (size_t)N * DIM;
    float* h3    = h2  + (size_t)N * DIM;
    float* hpool = h3  + (size_t)N * DIM;

    const long aggN  = (long)N * DIM;
    const long poolN = (long)NG * DIM;
    const int edgeTiles = (E + 15) / 16;
    const int nodeTiles = (N + 15) / 16;

    dim3 blk256(256), blk64(64);
    dim3 edgeGrid((unsigned)((edgeTiles + 7) / 8));
    dim3 nodeGrid((unsigned)((nodeTiles + 7) / 8));
    dim3 headGrid((unsigned)((nodeTiles + 1) / 2));
    dim3 zeroAggGrid((unsigned)((aggN + 255) / 256));
    dim3 zeroPoolGrid((unsigned)((poolN + 255) / 256));
    dim3 poolGrid((unsigned)((aggN + 255) / 256));

    const float* xin[3]  = {x, h1, h2};
    float*       hout[3] = {h1, h2, h3};

    for (int l = 0; l < 3; ++l) {
        zero_kernel<<<zeroAggGrid, blk256, 0, stream>>>(agg, aggN);
        gine_edge_kernel<<<edgeGrid, blk256, 0, stream>>>(xin[l], src, dst, eattr,
                                                          We[l], be[l], agg, E);
        gine_node_kernel<<<nodeGrid, blk256, 0, stream>>>(xin[l], agg, W[l], b[l],
                                                          g[l], bt[l], mn[l], vr[l],
                                                          hout[l], N);
    }
    zero_kernel<<<zeroPoolGrid, blk256, 0, stream>>>(hpool, poolN);
    pool_kernel<<<poolGrid, blk256, 0, stream>>>(h3, batch, hpool, N);
    head_kernel<<<headGrid, blk64, 0, stream>>>(h1, h2, h3, hpool, batch,
                                                Wl1, bl1, Wl2, bl2, (float*)d_out, N);
}